// AccS_82386062672504
// MI455X (gfx1250) — compile-verified
//
#include <hip/hip_runtime.h>
#include <stdint.h>

#define NUM_CLASSES 392
#define BATCH 262144
#define ROWS_PER_BLOCK 256
#define TILE_W 56          // floats per column tile; 392 = 7 * 56
#define N_TILES 7
#define LDS_STRIDE_F 60    // padded LDS row stride in floats (240B, 16B aligned)
#define CHUNKS_PER_ROW 14  // 56 floats = 14 float4 chunks

// Branchless insertion of v into descending sorted top-6 register file.
__device__ __forceinline__ void ins6(float v, float& t0, float& t1, float& t2,
                                     float& t3, float& t4, float& t5) {
  float a;
  a = fmaxf(t0, v); v = fminf(t0, v); t0 = a;
  a = fmaxf(t1, v); v = fminf(t1, v); t1 = a;
  a = fmaxf(t2, v); v = fminf(t2, v); t2 = a;
  a = fmaxf(t3, v); v = fminf(t3, v); t3 = a;
  a = fmaxf(t4, v); v = fminf(t4, v); t4 = a;
  t5 = fmaxf(t5, v);
}

__global__ __launch_bounds__(ROWS_PER_BLOCK) void topk_iou_kernel(
    const float* __restrict__ prob, const int* __restrict__ label,
    float* __restrict__ partial) {
  __shared__ __align__(16) float smem[ROWS_PER_BLOCK * LDS_STRIDE_F];  // 60 KB
  const int tid = threadIdx.x;
  const int row0 = blockIdx.x * ROWS_PER_BLOCK;
  const int row = row0 + tid;

  // Low 32 bits of the flat shared-aperture address == LDS byte offset.
  const unsigned lds_base = (unsigned)(uintptr_t)&smem[0];
  const uint64_t gbase = (uint64_t)(uintptr_t)prob;

  const int lab = label[row];  // coalesced; hoisted before the streaming loop

  float t0, t1, t2, t3, t4, t5;
  t0 = t1 = t2 = t3 = t4 = t5 = -INFINITY;
  float pl = 0.f;  // prob[row, lab]; captured from LDS in exactly one tile

  for (int tile = 0; tile < N_TILES; ++tile) {
    const int c0 = tile * TILE_W;
    if (tile > 0) __syncthreads();  // previous tile fully consumed before overwrite

    // Coalesced async copy: 3584 16B chunks per tile, 14 per thread.
    // Each wave instruction moves 32 lanes x 16B = 512B.
#pragma unroll
    for (int k = 0; k < CHUNKS_PER_ROW; ++k) {
      const int q = k * ROWS_PER_BLOCK + tid;          // 0..3583
      const int cr = q / CHUNKS_PER_ROW;               // tile row 0..255
      const int cc = q - cr * CHUNKS_PER_ROW;          // chunk col 0..13
      const uint64_t gaddr =
          gbase + ((uint64_t)(row0 + cr) * NUM_CLASSES + (unsigned)(c0 + cc * 4)) * 4ull;
      const unsigned laddr = lds_base + (unsigned)(cr * (LDS_STRIDE_F * 4) + cc * 16);
      asm volatile("global_load_async_to_lds_b128 %0, %1, off"
                   :: "v"(laddr), "v"(gaddr)
                   : "memory");
    }
    asm volatile("s_wait_asynccnt 0" ::: "memory");  // my wave's copies landed in LDS
    __syncthreads();                                 // everyone's copies landed

    // Capture the labeled element from LDS (removes the scattered global
    // gather; those lines are often L2-evicted by the 411MB stream).
    {
      const unsigned d = (unsigned)(lab - c0);
      if (d < TILE_W) pl = smem[tid * LDS_STRIDE_F + (int)d];
    }

    // Each thread scans its own row slice out of LDS.
    const float4* rowp = reinterpret_cast<const float4*>(&smem[tid * LDS_STRIDE_F]);
#pragma unroll
    for (int k = 0; k < CHUNKS_PER_ROW; ++k) {
      const float4 v = rowp[k];
      const float gm = fmaxf(fmaxf(v.x, v.y), fmaxf(v.z, v.w));
      if (gm > t5) {  // cheap filter; insertion itself is a no-op for small values
        ins6(v.x, t0, t1, t2, t3, t4, t5);
        ins6(v.y, t0, t1, t2, t3, t4, t5);
        ins6(v.z, t0, t1, t2, t3, t4, t5);
        ins6(v.w, t0, t1, t2, t3, t4, t5);
      }
    }
  }

  // thresh = 6th largest; count = #(strictly greater), all of which live in top-5.
  const float thresh = t5;
  const int count = (int)(t0 > thresh) + (int)(t1 > thresh) + (int)(t2 > thresh) +
                    (int)(t3 > thresh) + (int)(t4 > thresh);
  // inter = pl > thresh; union = count + 1 - inter  =>  iou = inter ? 1/count : 0
  float iou = (pl > thresh) ? (1.0f / (float)count) : 0.0f;

  // wave32 reduction, then per-block partial (deterministic, no atomics)
#pragma unroll
  for (int off = 16; off > 0; off >>= 1) iou += __shfl_down(iou, off, 32);

  __syncthreads();  // done reading smem as tile buffer
  if ((tid & 31) == 0) smem[tid >> 5] = iou;
  __syncthreads();
  if (tid == 0) {
    float s = 0.f;
#pragma unroll
    for (int w = 0; w < ROWS_PER_BLOCK / 32; ++w) s += smem[w];
    partial[blockIdx.x] = s;
  }
}

__global__ __launch_bounds__(256) void finalize_kernel(
    const float* __restrict__ partial, int n, float* __restrict__ out) {
  __shared__ float red[256];
  float s = 0.f;
  for (int i = threadIdx.x; i < n; i += 256) s += partial[i];
  red[threadIdx.x] = s;
  __syncthreads();
  for (int stride = 128; stride > 0; stride >>= 1) {
    if ((int)threadIdx.x < stride) red[threadIdx.x] += red[threadIdx.x + stride];
    __syncthreads();
  }
  if (threadIdx.x == 0) out[0] = red[0] * (1.0f / (float)BATCH);
}

extern "C" void kernel_launch(void* const* d_in, const int* in_sizes, int n_in,
                              void* d_out, int out_size, void* d_ws, size_t ws_size,
                              hipStream_t stream) {
  const float* prob = (const float*)d_in[0];
  const int* label = (const int*)d_in[1];  // JAX default: int64 demoted to int32
  float* partial = (float*)d_ws;           // 1024 floats of scratch
  const int nblocks = BATCH / ROWS_PER_BLOCK;  // 1024

  topk_iou_kernel<<<nblocks, ROWS_PER_BLOCK, 0, stream>>>(prob, label, partial);
  finalize_kernel<<<1, 256, 0, stream>>>(partial, nblocks, (float*)d_out);
}